// FGL_2138893714004
// MI455X (gfx1250) — compile-verified
//
#include <hip/hip_runtime.h>

// Problem constants (from reference)
constexpr int kN    = 8;
constexpr int kINC  = 32;
constexpr int kINN  = 50000;
constexpr int kOUTC = 64;
constexpr int kOUTN = 8192;
constexpr int kD    = 32;

typedef __attribute__((ext_vector_type(2))) float v2f;
typedef __attribute__((ext_vector_type(8))) float v8f;

// One block: batch n (blockIdx.y) x 16 output positions (blockIdx.x*16).
// 128 threads = 4 waves (wave32). Waves 0..3 each own a 16-wide k-tile.
__global__ __launch_bounds__(128) void fused_gather_max_wmma(
    const float* __restrict__ x,     // (N, INC, INN)
    const float* __restrict__ nf,    // (INC, INN)
    const float* __restrict__ ft,    // (INC, OUTC)
    const float* __restrict__ bias,  // (OUTC, OUTN)
    const int*   __restrict__ A,     // (OUTN, D)
    float*       __restrict__ out)   // (N, OUTC, OUTN)
{
  __shared__ int   sA[16 * kD];     // indices for the 16 'o' rows of this block
  __shared__ float sred[16 * 33];   // red[o][i], row stride 33 (bank-spread)

  const int tid    = threadIdx.x;
  const int o_base = blockIdx.x * 16;
  const int n      = blockIdx.y;

  // ---- Phase 0: stage the 16x32 tile of gather indices into LDS ----
  for (int j = tid; j < 16 * kD; j += 128)
    sA[j] = A[o_base * kD + j];
  __syncthreads();

  // ---- Phase 1: red[o][i] = max_d x[n,i,A[o,d]] * nf[i,A[o,d]] ----
  // Thread (i = tid&31, og = tid>>5) handles o_loc = og, og+4, og+8, og+12.
  {
    const int i  = tid & 31;
    const int og = tid >> 5;
    const float* __restrict__ xrow  = x  + ((size_t)n * kINC + i) * kINN;
    const float* __restrict__ nfrow = nf + (size_t)i * kINN;
    #pragma unroll
    for (int rep = 0; rep < 4; ++rep) {
      const int o_loc = og + rep * 4;
      const int* __restrict__ idxs = &sA[o_loc * kD];
      float m = -__builtin_inff();
      #pragma unroll 4
      for (int d = 0; d < kD; ++d) {
        const int idx = idxs[d];
        m = fmaxf(m, xrow[idx] * nfrow[idx]);
      }
      sred[o_loc * 33 + i] = m;
    }
  }
  __syncthreads();

  // ---- Phase 2: per-wave 16x16 tile of C[k,o] = sum_i ft[i,k]*red[i,o] ----
  // K = i (32) done as 8 chained V_WMMA_F32_16X16X4_F32 steps.
  {
    const int lane  = tid & 31;
    const int wave  = tid >> 5;        // k-tile id, 0..3
    const int m16   = lane & 15;       // A-row (k_local) == B-col (o_local)
    const int hi    = lane >> 4;       // 0: K={0,1}+4s, 1: K={2,3}+4s
    const int kbase = wave * 16;

    v8f c = {};
    #pragma unroll
    for (int s = 0; s < 8; ++s) {
      const int ks = 4 * s + 2 * hi;   // K index held in .x; .y holds ks+1
      v2f a, b;
      // A tile = ft^T : A[M=k_local, K=i] = ft[i*OUTC + kbase + m16]
      a.x = ft[(size_t)ks       * kOUTC + kbase + m16];
      a.y = ft[(size_t)(ks + 1) * kOUTC + kbase + m16];
      // B tile = red : B[K=i, N=o_local] = sred[o_local*33 + i]
      b.x = sred[m16 * 33 + ks];
      b.y = sred[m16 * 33 + ks + 1];
      // 8 args: (neg_a, A, neg_b, B, c_mod, C, reuse_a, reuse_b)
      c = __builtin_amdgcn_wmma_f32_16x16x4_f32(
              false, a, false, b, (short)0, c, false, false);
    }

    // ---- Phase 3: bias add + store ----
    // D layout: VGPR v -> row M = v + 8*hi (= k_local), col N = m16 (= o_local)
    #pragma unroll
    for (int v = 0; v < 8; ++v) {
      const int k = kbase + v + 8 * hi;
      const int o = o_base + m16;
      const size_t bidx = (size_t)k * kOUTN + o;
      out[(size_t)n * kOUTC * kOUTN + bidx] = c[v] + bias[bidx];
    }
  }
}

extern "C" void kernel_launch(void* const* d_in, const int* in_sizes, int n_in,
                              void* d_out, int out_size, void* d_ws, size_t ws_size,
                              hipStream_t stream) {
  const float* x    = (const float*)d_in[0];
  const float* nf   = (const float*)d_in[1];
  const float* ft   = (const float*)d_in[2];
  const float* bias = (const float*)d_in[3];
  const int*   A    = (const int*)d_in[4];
  float* out = (float*)d_out;

  dim3 grid(kOUTN / 16, kN);   // 512 x 8 blocks
  fused_gather_max_wmma<<<grid, 128, 0, stream>>>(x, nf, ft, bias, A, out);

  (void)in_sizes; (void)n_in; (void)out_size; (void)d_ws; (void)ws_size;
}